// BCEDiceLossWeight_52604759442070
// MI455X (gfx1250) — compile-verified
//
#include <hip/hip_runtime.h>
#include <math.h>

// Tensor geometry from the reference: (B,C,D,H,W) = (4,3,128,128,128)
#define N_B 4
#define N_C 3
#define N_SLICE 2097152          // 128*128*128 elements per (b,c) slice
#define N_SLICES (N_B * N_C)     // 12
#define N_TOTAL (N_SLICES * (long long)N_SLICE) // 25,165,824
#define BLOCKS_PER_SLICE 64
#define THREADS 256

typedef __attribute__((ext_vector_type(2))) float v2f;
typedef __attribute__((ext_vector_type(4))) float v4f;
typedef __attribute__((ext_vector_type(8))) float v8f;

// ---------------------------------------------------------------------------
// Wave-level (32-lane) sum using V_WMMA_F32_16X16X4_F32 with a ones B-matrix.
// A layout (ISA 7.12.2, 32-bit A 16x4): lane m (m<16) VGPR0=A[m][0], VGPR1=A[m][1];
// lane m+16 VGPR0=A[m][2], VGPR1=A[m][3]. We put x in VGPR0 and 0 in VGPR1, so
// with B==1 everywhere: D[m][n] = x(lane m) + x(lane m+16) for every column n.
// C/D layout: lane L<16 holds D[0..7][L] in its 8 VGPRs, lane L+16 holds
// D[8..15][L]. So sum-of-8-VGPRs plus the lane^16 partner = full 32-lane sum.
// Must be called with EXEC all ones (uniform control flow).
// ---------------------------------------------------------------------------
__device__ __forceinline__ float wave_sum32(float x) {
#if defined(__AMDGCN__) && __has_builtin(__builtin_amdgcn_wmma_f32_16x16x4_f32)
  v2f a; a.x = x;    a.y = 0.0f;
  v2f b; b.x = 1.0f; b.y = 1.0f;
  v8f c = {};
  c = __builtin_amdgcn_wmma_f32_16x16x4_f32(
      /*neg_a=*/false, a, /*neg_b=*/false, b,
      /*c_mod=*/(short)0, c, /*reuse_a=*/false, /*reuse_b=*/false);
  float vs = c[0] + c[1] + c[2] + c[3] + c[4] + c[5] + c[6] + c[7];
  vs += __shfl_xor(vs, 16, 32);
  return vs;   // every lane holds the full 32-lane sum
#else
  for (int off = 16; off > 0; off >>= 1) x += __shfl_xor(x, off, 32);
  return x;
#endif
}

// ---------------------------------------------------------------------------
// Kernel 1: streaming partial sums. blockIdx.y = slice (b*3+c).
// 6 accumulators: {sum(p*t), sum(p), sum(t), sum(t*pred), sum(pred), sum(bce)}
// ---------------------------------------------------------------------------
__global__ void __launch_bounds__(THREADS)
bce_dice_partials(const float* __restrict__ logits,
                  const float* __restrict__ targets,
                  float* __restrict__ ws) {
  const int slice = blockIdx.y;
  const size_t base = (size_t)slice * (size_t)N_SLICE;
  const v4f* lg = (const v4f*)(logits + base);
  const v4f* tg = (const v4f*)(targets + base);

  const int nvec   = N_SLICE / 4;                   // 524288 float4 per slice
  const int stride = BLOCKS_PER_SLICE * THREADS;    // 16384 -> exactly 32 iters
  float s_pt = 0.f, s_p = 0.f, s_t = 0.f, s_tp = 0.f, s_pr = 0.f, s_bce = 0.f;

  for (int i = blockIdx.x * THREADS + threadIdx.x; i < nvec; i += stride) {
    v4f l4 = __builtin_nontemporal_load(&lg[i]);    // streamed once: TH=NT
    v4f t4 = __builtin_nontemporal_load(&tg[i]);
#pragma unroll
    for (int j = 0; j < 4; ++j) {
      float l = l4[j];
      float t = t4[j];
      float p = 1.0f / (1.0f + __expf(-l));         // sigmoid
      s_pt += p * t;
      s_p  += p;
      s_t  += t;
      float pred = (l >= 0.5f) ? 1.0f : 0.0f;       // ref thresholds raw logits
      s_tp += t * pred;
      s_pr += pred;
      // stable BCE-with-logits: max(l,0) - l*t + log1p(exp(-|l|))
      s_bce += fmaxf(l, 0.0f) - l * t + log1pf(__expf(-fabsf(l)));
    }
  }

  // Wave reduction via WMMA (uniform flow; EXEC all ones here).
  float r0 = wave_sum32(s_pt);
  float r1 = wave_sum32(s_p);
  float r2 = wave_sum32(s_t);
  float r3 = wave_sum32(s_tp);
  float r4 = wave_sum32(s_pr);
  float r5 = wave_sum32(s_bce);

  __shared__ float red[THREADS / 32][6];
  const int wave = threadIdx.x >> 5;
  const int lane = threadIdx.x & 31;
  if (lane == 0) {
    red[wave][0] = r0; red[wave][1] = r1; red[wave][2] = r2;
    red[wave][3] = r3; red[wave][4] = r4; red[wave][5] = r5;
  }
  __syncthreads();

  if (threadIdx.x < 6) {
    float acc = 0.f;
#pragma unroll
    for (int w = 0; w < THREADS / 32; ++w) acc += red[w][threadIdx.x];
    atomicAdd(&ws[slice * 6 + threadIdx.x], acc);   // global_atomic_add_f32
  }
}

// ---------------------------------------------------------------------------
// Kernel 2: tiny epilogue — combine the 12x6 partials exactly as the reference.
// ---------------------------------------------------------------------------
__global__ void bce_dice_finalize(const float* __restrict__ ws,
                                  float* __restrict__ out) {
  if (threadIdx.x != 0) return;
  const float EPS = 1e-9f;
  float g_pt = 0.f, g_p = 0.f, g_t = 0.f, bce = 0.f;
  float pc[N_C] = {0.f, 0.f, 0.f};
  for (int b = 0; b < N_B; ++b) {
    for (int c = 0; c < N_C; ++c) {
      const float* s = ws + (b * N_C + c) * 6;
      g_pt += s[0];
      g_p  += s[1];
      g_t  += s[2];
      bce  += s[5];
      float tsum  = s[2];
      float psum  = s[4];
      float inter = 2.0f * s[3];
      float score = (tsum == 0.0f && psum == 0.0f)
                        ? 1.0f
                        : (inter + EPS) / (tsum + psum);
      pc[c] += score;
    }
  }
  const float invB = 1.0f / (float)N_B;
  float dice_loss = 1.0f - (2.0f * g_pt + EPS) / (g_p + g_t);
  float bce_mean  = bce / (float)N_TOTAL;
  out[0] = bce_mean + dice_loss * 0.5f
         + pc[0] * invB * 0.2f + pc[1] * invB * 0.1f + pc[2] * invB * 0.2f;
}

// ---------------------------------------------------------------------------
extern "C" void kernel_launch(void* const* d_in, const int* in_sizes, int n_in,
                              void* d_out, int out_size, void* d_ws, size_t ws_size,
                              hipStream_t stream) {
  const float* logits  = (const float*)d_in[0];
  const float* targets = (const float*)d_in[1];
  float* ws = (float*)d_ws;

  // 12 slices x 6 accumulators; capture-safe async memset each call.
  hipMemsetAsync(ws, 0, N_SLICES * 6 * sizeof(float), stream);

  dim3 grid(BLOCKS_PER_SLICE, N_SLICES);
  bce_dice_partials<<<grid, THREADS, 0, stream>>>(logits, targets, ws);
  bce_dice_finalize<<<1, 32, 0, stream>>>(ws, (float*)d_out);
}